// AdaptiveGNN_33552284516387
// MI455X (gfx1250) — compile-verified
//
#include <hip/hip_runtime.h>
#include <hip/hip_bf16.h>
#include <math.h>

typedef float v2f __attribute__((ext_vector_type(2)));
typedef float v8f __attribute__((ext_vector_type(8)));
typedef unsigned int u32x4 __attribute__((ext_vector_type(4)));
typedef int i32x4 __attribute__((ext_vector_type(4)));
typedef int i32x8 __attribute__((ext_vector_type(8)));

#define N_NODES 50000
#define KNBR 16
#define DH 128
#define NCLS 8
#define EPS_LN 1e-5f
#define CLAMP_E 1e-5f

#define GEMM_BLOCKS (N_NODES / 16)        // 3125, exact
#define AGG_BLOCKS  (N_NODES / 8)         // 6250, exact
#define PT_BLOCKS   ((N_NODES + 255) / 256)
#define STAT_BLOCKS 200
#define FIN_BLOCKS  1024

__device__ __forceinline__ float sigm(float x) { return 1.0f / (1.0f + expf(-x)); }

// ---------------------------------------------------------------------------
// Kernel 1: hx = feat @ W  (N x 128) @ (128 x 128), fp32 WMMA 16x16x4.
// Block = 16 rows x 128 cols of output. 8 waves; wave j owns N-tile j (16 cols).
// feat tile (16x128, 8KB) staged into LDS by the Tensor Data Mover (TDM):
// wave 0 issues one tensor_load_to_lds with a 2D D# (tile 16x128, f32),
// waits on TENSORcnt, then the workgroup barrier publishes the tile.
// ---------------------------------------------------------------------------
__global__ __launch_bounds__(256) void gnn_gemm_wmma_kernel(
    const float* __restrict__ feat, const float* __restrict__ W,
    float* __restrict__ hx) {
  __shared__ float atile[16 * DH];
  const int tid   = threadIdx.x;
  const int wave  = tid >> 5;
  const int lane  = tid & 31;
  const int rowblk = blockIdx.x * 16;

#if __has_builtin(__builtin_amdgcn_tensor_load_to_lds) && __has_builtin(__builtin_amdgcn_s_wait_tensorcnt)
  if (wave == 0) {
    // --- Tensor DMA descriptor (ISA cdna5 §8.3/8.4), 2D tile 16 rows x 128 f32.
    const unsigned lds_off = (unsigned)(uintptr_t)(void*)atile;     // LDS byte offset
    const size_t ga = (size_t)(const void*)(feat + (size_t)rowblk * DH);
    u32x4 g0;
    g0[0] = 1u;                                        // count=1 (valid user D#)
    g0[1] = lds_off;                                   // lds_addr [63:32]
    g0[2] = (unsigned)(ga & 0xFFFFFFFFu);              // global_addr [95:64]
    g0[3] = (unsigned)((ga >> 32) & 0x01FFFFFFu)       // global_addr [120:96]
          | 0x80000000u;                               // type=2 ("image") [127:126]
    i32x8 g1;
    g1[0] = 0x00020000;        // workgroup_mask=0 (no cluster), data_size=2 -> 4B
    g1[1] = (int)(128u << 16); // tensor_dim0 = 128 (bits 79:48, low half here)
    g1[2] = (int)(16u << 16);  // tensor_dim1 = 16  (bits 111:80, low half here)
    g1[3] = (int)(128u << 16); // tile_dim0 = 128   (bits 127:112)
    g1[4] = 16;                // tile_dim1 = 16    (bits 143:128); tile_dim2=0
    g1[5] = 128;               // tensor_dim0_stride = 128 (bits 207:160)
    g1[6] = 0;                 // stride high bits / tensor_dim1_stride
    g1[7] = 0;
    i32x4 gz4 = {};            // groups 2/3: 2D tile -> all-zero (dims unused)
    i32x8 gz8 = {};            // trailing group (6-arg toolchain form): zero
    __builtin_amdgcn_tensor_load_to_lds(g0, g1, gz4, gz4, gz8, 0);
    __builtin_amdgcn_s_wait_tensorcnt(0);              // s_wait_tensorcnt 0
  }
  __syncthreads();
#else
  // Fallback: coalesced manual staging (512 float4 across 256 threads).
  {
    const float4* src = (const float4*)(feat + (size_t)rowblk * DH);
    float4* dst = (float4*)atile;
    dst[tid]       = src[tid];
    dst[tid + 256] = src[tid + 256];
  }
  __syncthreads();
#endif

  const int half = lane >> 4;       // 0: lanes 0-15, 1: lanes 16-31
  const int l16  = lane & 15;
  const int colg = wave * 16 + l16; // output column for B/C striping

  v8f c = {};
  #pragma unroll 4
  for (int k0 = 0; k0 < DH; k0 += 4) {
    // A 16x4 f32: VGPR0 = K{k0 | k0+2}, VGPR1 = K{k0+1 | k0+3}, M = lane%16
    const int kA = k0 + half * 2;
    v2f a, b;
    a.x = atile[l16 * DH + kA];
    a.y = atile[l16 * DH + kA + 1];
    // B 4x16 f32: rows striped across lanes (lanes0-15 -> K=kA, same pairing)
    b.x = W[(size_t)kA * DH + colg];
    b.y = W[(size_t)(kA + 1) * DH + colg];
    c = __builtin_amdgcn_wmma_f32_16x16x4_f32(false, a, false, b,
                                              (short)0, c, false, false);
  }
  // C/D layout: VGPR r -> M = r (lanes 0-15) / M = r+8 (lanes 16-31), N = lane%16
  #pragma unroll
  for (int r = 0; r < 8; ++r) {
    const int m = r + half * 8;
    hx[(size_t)(rowblk + m) * DH + colg] = c[r];
  }
}

// ---------------------------------------------------------------------------
// Kernel 2: h[i] = b + sum_k hx[nbr[i,k]]   (gathers are L2-resident: 25.6MB)
// One wave per node, one float4 per lane.
// ---------------------------------------------------------------------------
__global__ __launch_bounds__(256) void gnn_agg_bias_kernel(
    const float* __restrict__ hx, const int* __restrict__ nbr,
    const float* __restrict__ bias, float* __restrict__ h) {
  const int node = blockIdx.x * 8 + (threadIdx.x >> 5);
  const int lane = threadIdx.x & 31;
  if (node >= N_NODES) return;
  float4 acc = ((const float4*)bias)[lane];
  const int* nb = nbr + (size_t)node * KNBR;
  #pragma unroll
  for (int k = 0; k < KNBR; ++k) {
    float4 v = ((const float4*)(hx + (size_t)nb[k] * DH))[lane];
    acc.x += v.x; acc.y += v.y; acc.z += v.z; acc.w += v.w;
  }
  ((float4*)(h + (size_t)node * DH))[lane] = acc;
}

// ---------------------------------------------------------------------------
// Kernel 3a: labels[i] = argmax_c (h[i] @ Wy[:,c] + by[c]).
// (softmax is monotone -> skip it; logits only feed argmax in the reference)
// ---------------------------------------------------------------------------
__global__ __launch_bounds__(256) void gnn_argmax_kernel(
    const float* __restrict__ h, const float* __restrict__ Wy,
    const float* __restrict__ by, int* __restrict__ labels) {
  const int i = blockIdx.x * 256 + threadIdx.x;
  if (i >= N_NODES) return;
  float acc[NCLS];
  #pragma unroll
  for (int c = 0; c < NCLS; ++c) acc[c] = by[c];
  const float* hr = h + (size_t)i * DH;
  for (int d = 0; d < DH; ++d) {
    const float hv = hr[d];
    const float* wr = Wy + (size_t)d * NCLS;
    #pragma unroll
    for (int c = 0; c < NCLS; ++c) acc[c] += hv * wr[c];
  }
  int best = 0; float bv = acc[0];
  #pragma unroll
  for (int c = 1; c < NCLS; ++c)
    if (acc[c] > bv) { bv = acc[c]; best = c; }   // first-max tie rule
  labels[i] = best;
}

// ---------------------------------------------------------------------------
// Kernel 3b: f1 = #(neighbor label == center), f2 = -sum clip(hist)*log(hist)
// ---------------------------------------------------------------------------
__global__ __launch_bounds__(256) void gnn_f1f2_kernel(
    const int* __restrict__ labels, const int* __restrict__ nbr,
    float* __restrict__ f1o, float* __restrict__ f2o) {
  const int i = blockIdx.x * 256 + threadIdx.x;
  if (i >= N_NODES) return;
  int cnt[NCLS];
  #pragma unroll
  for (int c = 0; c < NCLS; ++c) cnt[c] = 0;
  const int center = labels[i];
  const int* nb = nbr + (size_t)i * KNBR;
  int eq = 0;
  #pragma unroll
  for (int k = 0; k < KNBR; ++k) {
    const int p = labels[nb[k]];
    cnt[p]++;
    eq += (p == center) ? 1 : 0;
  }
  f1o[i] = (float)eq;
  float f2 = 0.0f;
  #pragma unroll
  for (int c = 0; c < NCLS; ++c) {
    const float cv = (cnt[c] > 0) ? (float)cnt[c] : CLAMP_E;
    f2 -= cv * logf(cv);
  }
  f2o[i] = f2;
}

// ---------------------------------------------------------------------------
// Kernel 3c/3d: deterministic two-stage global LayerNorm stats (no atomics).
// ---------------------------------------------------------------------------
__global__ __launch_bounds__(256) void gnn_stats_partial_kernel(
    const float* __restrict__ f1, const float* __restrict__ f2,
    float* __restrict__ part) {
  __shared__ float sh[4][256];
  const int t = threadIdx.x;
  float s1 = 0.f, s1q = 0.f, s2 = 0.f, s2q = 0.f;
  for (int i = blockIdx.x * 256 + t; i < N_NODES; i += gridDim.x * 256) {
    const float a = f1[i], b = f2[i];
    s1 += a; s1q += a * a; s2 += b; s2q += b * b;
  }
  sh[0][t] = s1; sh[1][t] = s1q; sh[2][t] = s2; sh[3][t] = s2q;
  __syncthreads();
  for (int st = 128; st > 0; st >>= 1) {
    if (t < st) {
      sh[0][t] += sh[0][t + st]; sh[1][t] += sh[1][t + st];
      sh[2][t] += sh[2][t + st]; sh[3][t] += sh[3][t + st];
    }
    __syncthreads();
  }
  if (t == 0) {
    part[blockIdx.x * 4 + 0] = sh[0][0];
    part[blockIdx.x * 4 + 1] = sh[1][0];
    part[blockIdx.x * 4 + 2] = sh[2][0];
    part[blockIdx.x * 4 + 3] = sh[3][0];
  }
}

__global__ __launch_bounds__(256) void gnn_stats_final_kernel(
    const float* __restrict__ part, float* __restrict__ red) {
  __shared__ float sh[256];
  const int t = threadIdx.x;
  const int j = t & 3, b0 = t >> 2;
  float s = 0.f;
  for (int b = b0; b < STAT_BLOCKS; b += 64) s += part[b * 4 + j];
  sh[t] = s;
  __syncthreads();
  for (int st = 128; st >= 4; st >>= 1) {
    if (t < st) sh[t] += sh[t + st];
    __syncthreads();
  }
  if (t == 0) {
    const float invN = 1.0f / (float)N_NODES;
    const float m1 = sh[0] * invN;
    const float v1 = sh[1] * invN - m1 * m1;
    const float m2 = sh[2] * invN;
    const float v2 = sh[3] * invN - m2 * m2;
    red[0] = m1; red[1] = rsqrtf(v1 + EPS_LN);
    red[2] = m2; red[3] = rsqrtf(v2 + EPS_LN);
  }
}

// ---------------------------------------------------------------------------
// Kernel 3e: z, gate=min(old_z,z), hnext = h + gate * sum_k h[nbr[k]].
// One wave per node; z computed redundantly per lane (cheap, no shuffle dep).
// ---------------------------------------------------------------------------
__global__ __launch_bounds__(256) void gnn_gate_update_kernel(
    const float* __restrict__ hcur, const int* __restrict__ nbr,
    const float* __restrict__ f1, const float* __restrict__ f2,
    const float* __restrict__ red, const float* __restrict__ tau1,
    const float* __restrict__ tau2, const float* __restrict__ zold,
    float* __restrict__ znew, float* __restrict__ hnext, int first) {
  const int node = blockIdx.x * 8 + (threadIdx.x >> 5);
  const int lane = threadIdx.x & 31;
  if (node >= N_NODES) return;
  const float nf1 = (f1[node] - red[0]) * red[1];
  const float nf2 = (f2[node] - red[2]) * red[3];
  const float z = sigm(-(nf1 - tau1[0])) * sigm(-(nf2 - tau2[0]));
  const float oz = first ? 1.0f : zold[node];
  const float gate = fminf(oz, z);
  if (lane == 0) znew[node] = z;
  const int* nb = nbr + (size_t)node * KNBR;
  float4 agg = make_float4(0.f, 0.f, 0.f, 0.f);
  #pragma unroll
  for (int k = 0; k < KNBR; ++k) {
    float4 v = ((const float4*)(hcur + (size_t)nb[k] * DH))[lane];
    agg.x += v.x; agg.y += v.y; agg.z += v.z; agg.w += v.w;
  }
  float4 hv = ((const float4*)(hcur + (size_t)node * DH))[lane];
  hv.x += gate * agg.x; hv.y += gate * agg.y;
  hv.z += gate * agg.z; hv.w += gate * agg.w;
  ((float4*)(hnext + (size_t)node * DH))[lane] = hv;
}

// ---------------------------------------------------------------------------
// Kernel 4: out[c] = sum_e h.flat[e] * Wc[e,c] + bc[c].
// Wc = 204.8MB: the HBM-bound stream of this workload. Fully coalesced,
// grid-stride, per-block partials -> deterministic final reduce.
// ---------------------------------------------------------------------------
__global__ __launch_bounds__(256) void gnn_final_partial_kernel(
    const float* __restrict__ h, const float* __restrict__ Wc,
    float* __restrict__ gpart) {
  float acc[NCLS];
  #pragma unroll
  for (int c = 0; c < NCLS; ++c) acc[c] = 0.f;
  const size_t total  = (size_t)N_NODES * DH;
  const size_t stride = (size_t)gridDim.x * 256;
  for (size_t e = (size_t)blockIdx.x * 256 + threadIdx.x; e < total; e += stride) {
    const float v = h[e];
    const float* w = Wc + e * NCLS;
    __builtin_prefetch(w + stride * NCLS, 0, 0);   // global_prefetch_b8, next stripe
    #pragma unroll
    for (int c = 0; c < NCLS; ++c) acc[c] += v * w[c];
  }
  __shared__ float sh[256 * NCLS];
  #pragma unroll
  for (int c = 0; c < NCLS; ++c) sh[threadIdx.x * NCLS + c] = acc[c];
  __syncthreads();
  for (int st = 128; st > 0; st >>= 1) {
    if (threadIdx.x < st) {
      #pragma unroll
      for (int c = 0; c < NCLS; ++c)
        sh[threadIdx.x * NCLS + c] += sh[(threadIdx.x + st) * NCLS + c];
    }
    __syncthreads();
  }
  if (threadIdx.x < NCLS) gpart[blockIdx.x * NCLS + threadIdx.x] = sh[threadIdx.x];
}

__global__ __launch_bounds__(256) void gnn_final_reduce_kernel(
    const float* __restrict__ gpart, const float* __restrict__ bc,
    float* __restrict__ out) {
  __shared__ float sh[256];
  const int t = threadIdx.x;
  const int c = t & 7, b0 = t >> 3;
  float s = 0.f;
  for (int b = b0; b < FIN_BLOCKS; b += 32) s += gpart[b * NCLS + c];
  sh[t] = s;
  __syncthreads();
  for (int st = 128; st >= 8; st >>= 1) {   // stride stays multiple of 8
    if (t < st) sh[t] += sh[t + st];
    __syncthreads();
  }
  if (t < NCLS) out[t] = sh[t] + bc[t];
}

// ---------------------------------------------------------------------------
extern "C" void kernel_launch(void* const* d_in, const int* in_sizes, int n_in,
                              void* d_out, int out_size, void* d_ws, size_t ws_size,
                              hipStream_t stream) {
  const float* feat = (const float*)d_in[0];
  const float* Wgcn = (const float*)d_in[1];
  const float* bgcn = (const float*)d_in[2];
  const float* Wy   = (const float*)d_in[3];
  const float* by   = (const float*)d_in[4];
  const float* tau1 = (const float*)d_in[5];
  const float* tau2 = (const float*)d_in[6];
  const float* Wc   = (const float*)d_in[7];
  const float* bc   = (const float*)d_in[8];
  const int*   nbr  = (const int*)d_in[9];
  float* out = (float*)d_out;

  char* ws = (char*)d_ws;
  size_t off = 0;
  auto wsAlloc = [&](size_t bytes) -> void* {
    void* p = (void*)(ws + off);
    off = (off + bytes + 255) & ~(size_t)255;
    return p;
  };
  float* bufA   = (float*)wsAlloc((size_t)N_NODES * DH * sizeof(float));
  float* bufB   = (float*)wsAlloc((size_t)N_NODES * DH * sizeof(float));
  float* f1     = (float*)wsAlloc((size_t)N_NODES * sizeof(float));
  float* f2     = (float*)wsAlloc((size_t)N_NODES * sizeof(float));
  float* z0     = (float*)wsAlloc((size_t)N_NODES * sizeof(float));
  float* z1     = (float*)wsAlloc((size_t)N_NODES * sizeof(float));
  int*   labels = (int*)wsAlloc((size_t)N_NODES * sizeof(int));
  float* part   = (float*)wsAlloc((size_t)STAT_BLOCKS * 4 * sizeof(float));
  float* red    = (float*)wsAlloc(4 * sizeof(float));
  float* gpart  = (float*)wsAlloc((size_t)FIN_BLOCKS * NCLS * sizeof(float));

  // Layer 0: WMMA GEMM (TDM-staged tiles) into bufA, then gather-sum + bias.
  gnn_gemm_wmma_kernel<<<GEMM_BLOCKS, 256, 0, stream>>>(feat, Wgcn, bufA);
  gnn_agg_bias_kernel<<<AGG_BLOCKS, 256, 0, stream>>>(bufA, nbr, bgcn, bufB);

  float* h  = bufB;
  float* hn = bufA;
  float* za = z0;
  float* zb = z1;
  for (int it = 0; it < 3; ++it) {
    gnn_argmax_kernel<<<PT_BLOCKS, 256, 0, stream>>>(h, Wy, by, labels);
    gnn_f1f2_kernel<<<PT_BLOCKS, 256, 0, stream>>>(labels, nbr, f1, f2);
    gnn_stats_partial_kernel<<<STAT_BLOCKS, 256, 0, stream>>>(f1, f2, part);
    gnn_stats_final_kernel<<<1, 256, 0, stream>>>(part, red);
    gnn_gate_update_kernel<<<AGG_BLOCKS, 256, 0, stream>>>(
        h, nbr, f1, f2, red, tau1, tau2, za, zb, hn, it == 0 ? 1 : 0);
    float* t = h; h = hn; hn = t;
    t = za; za = zb; zb = t;
  }

  gnn_final_partial_kernel<<<FIN_BLOCKS, 256, 0, stream>>>(h, Wc, gpart);
  gnn_final_reduce_kernel<<<1, 256, 0, stream>>>(gpart, bc, out);
}